// QuantizedBn2d_64115271794718
// MI455X (gfx1250) — compile-verified
//
#include <hip/hip_runtime.h>

typedef float v4f __attribute__((ext_vector_type(4)));

// Requantization core: sub = x*w + c0; gemmlowp rounding-doubling high mul by
// Mk (always in [2^30,2^31) -> positive, so nudge sign == sign(sub2));
// round-half-away-from-zero right shift folded to add-before-shift with
// block-uniform constants; zero-point add; int8 clamp.
__device__ __forceinline__ v4f qbn_apply(v4f v, int wv, int c0, long long Mk64,
                                         int pre, int post, int rnd_pos,
                                         int rnd_neg, int zk)
{
    v4f r;
#pragma unroll
    for (int j = 0; j < 4; ++j) {
        int xi   = (int)v[j];                       // x holds exact ints
        int sub  = xi * wv + c0;                    // one v_mad_i32
        int sub2 = (int)((unsigned)sub << pre);     // |sub| < 2^15, pre <= 2
        long long prod   = (long long)sub2 * Mk64;  // v_mul_lo + v_mul_hi
        long long nudged = prod + (sub2 >= 0 ? (1LL << 30)
                                             : (1LL - (1LL << 30)));
        int mul = (int)(nudged >> 31);              // fits in ~17 bits
        int rr  = mul + (mul < 0 ? rnd_neg : rnd_pos);
        int t   = (rr >> post) + zk;
        t = t < -128 ? -128 : (t > 127 ? 127 : t);  // v_med3-able clamp
        r[j] = (float)t;
    }
    return r;
}

// One block per (b, c): grid = (C, B), so b/c come straight from blockIdx
// (no integer division).  All parameters are block-uniform -> SGPR path.
// NV = vec4 elements per (b,c) image plane, known at compile time so the
// whole strip is load-batched: FULL unconditional b128 loads issued
// back-to-back (deep MLP), then compute, then NT stores.
template <int NV>
__global__ __launch_bounds__(256) void qbn_pcq_fixed(
    const float* __restrict__ x,
    const int*   __restrict__ bc,
    const int*   __restrict__ z3,
    const int*   __restrict__ M0,
    const int*   __restrict__ shift,
    const int*   __restrict__ mean,
    const int*   __restrict__ wds,
    const int*   __restrict__ bias,
    float*       __restrict__ out)
{
    const int c = blockIdx.x;
    const int b = blockIdx.y;
    const int C = gridDim.x;

    // ---- block-uniform parameters (scalar loads / SALU) ----
    const int k   = bc[b];
    const int wv  = wds[k * C + c];
    const int c0  = bias[k * C + c] - mean[k * C + c] * wv;
    const int s   = shift[k];
    const int zk  = z3[k];
    const long long Mk64 = (long long)M0[k];        // in [2^30, 2^31): > 0
    const int pre  = (s < 0) ? -s : 0;
    const int post = (s > 0) ?  s : 0;
    const int rnd_pos = post ? (1 << (post - 1))       : 0;
    const int rnd_neg = post ? ((1 << (post - 1)) - 1) : 0;

    const size_t base = (size_t)(b * C + c) * NV;
    const v4f* __restrict__ xin = (const v4f*)x + base;
    v4f*       __restrict__ op  = (v4f*)out      + base;

    constexpr int FULL = NV / 256;                  // 3 for NV = 784
    constexpr int TAIL = NV % 256;                  // 16 for NV = 784
    const int t0 = threadIdx.x;

    v4f v[FULL + (TAIL ? 1 : 0)];
#pragma unroll
    for (int u = 0; u < FULL; ++u)                  // batched NT b128 loads
        v[u] = __builtin_nontemporal_load(xin + t0 + u * 256);
    if constexpr (TAIL > 0) {
        if (t0 < TAIL)
            v[FULL] = __builtin_nontemporal_load(xin + t0 + FULL * 256);
    }

#pragma unroll
    for (int u = 0; u < FULL; ++u) {
        v4f r = qbn_apply(v[u], wv, c0, Mk64, pre, post, rnd_pos, rnd_neg, zk);
        __builtin_nontemporal_store(r, op + t0 + u * 256);
    }
    if constexpr (TAIL > 0) {
        if (t0 < TAIL) {
            v4f r = qbn_apply(v[FULL], wv, c0, Mk64, pre, post, rnd_pos,
                              rnd_neg, zk);
            __builtin_nontemporal_store(r, op + t0 + FULL * 256);
        }
    }
}

// Generic fallback for unexpected shapes (runtime loop + prefetch).
__global__ __launch_bounds__(256) void qbn_pcq_generic(
    const float* __restrict__ x,
    const int*   __restrict__ bc,
    const int*   __restrict__ z3,
    const int*   __restrict__ M0,
    const int*   __restrict__ shift,
    const int*   __restrict__ mean,
    const int*   __restrict__ wds,
    const int*   __restrict__ bias,
    float*       __restrict__ out,
    int HW4)
{
    const int c = blockIdx.x;
    const int b = blockIdx.y;
    const int C = gridDim.x;

    const int k   = bc[b];
    const int wv  = wds[k * C + c];
    const int c0  = bias[k * C + c] - mean[k * C + c] * wv;
    const int s   = shift[k];
    const int zk  = z3[k];
    const long long Mk64 = (long long)M0[k];
    const int pre  = (s < 0) ? -s : 0;
    const int post = (s > 0) ?  s : 0;
    const int rnd_pos = post ? (1 << (post - 1))       : 0;
    const int rnd_neg = post ? ((1 << (post - 1)) - 1) : 0;

    const size_t base = (size_t)(b * C + c) * HW4;
    const v4f* __restrict__ xin = (const v4f*)x + base;
    v4f*       __restrict__ op  = (v4f*)out      + base;

    for (int i = threadIdx.x; i < HW4; i += 256) {
        v4f v = __builtin_nontemporal_load(xin + i);
        __builtin_prefetch((const void*)(xin + i + 256), 0, 0); // global_prefetch_b8
        v4f r = qbn_apply(v, wv, c0, Mk64, pre, post, rnd_pos, rnd_neg, zk);
        __builtin_nontemporal_store(r, op + i);
    }
}

extern "C" void kernel_launch(void* const* d_in, const int* in_sizes, int n_in,
                              void* d_out, int out_size, void* d_ws, size_t ws_size,
                              hipStream_t stream) {
    (void)n_in; (void)out_size; (void)d_ws; (void)ws_size;

    const float* x    = (const float*)d_in[0];
    const int*  bc    = (const int*)d_in[1];
    const int*  z3    = (const int*)d_in[2];
    const int*  M0    = (const int*)d_in[3];
    const int*  shift = (const int*)d_in[4];
    const int*  mean  = (const int*)d_in[5];
    const int*  wds   = (const int*)d_in[6];
    const int*  bias  = (const int*)d_in[7];
    float*      out   = (float*)d_out;

    const int B   = in_sizes[1];                   // 32
    const int K   = in_sizes[2];                   // 8
    const int C   = in_sizes[5] / K;               // 256
    const long long n = (long long)in_sizes[0];    // B*C*H*W
    const int HW  = (int)(n / ((long long)B * C)); // 3136
    const int HW4 = HW / 4;                        // 784 vec4 per (b,c)

    dim3 grid((unsigned)C, (unsigned)B);           // 256 x 32 = 8192 blocks
    if (HW4 == 784) {
        qbn_pcq_fixed<784><<<grid, 256, 0, stream>>>(x, bc, z3, M0, shift,
                                                     mean, wds, bias, out);
    } else {
        qbn_pcq_generic<<<grid, 256, 0, stream>>>(x, bc, z3, M0, shift,
                                                  mean, wds, bias, out, HW4);
    }
}